// YOLOLayer_26800595927562
// MI455X (gfx1250) — compile-verified
//
#include <hip/hip_runtime.h>
#include <stdint.h>

#define B_      16
#define A_      3
#define C_      80
#define H_      52
#define W_      52
#define HW_     (H_ * W_)           // 2704
#define CELLS_  (B_ * A_ * HW_)     // 129792
#define CH_     85
#define NT_     256
#define STRIDE_ 8.0f
#define TILE_   128                  // cells per block
#define ROW_    132                  // padded LDS row (TDM pad: 4 DW per 128 DW)

typedef __attribute__((ext_vector_type(4))) unsigned v4u;
typedef __attribute__((ext_vector_type(8))) int      v8i;
typedef __attribute__((ext_vector_type(4))) int      v4i;

__device__ __constant__ float d_aw[3] = {10.f, 16.f, 33.f};
__device__ __constant__ float d_ah[3] = {13.f, 30.f, 23.f};

__device__ __forceinline__ float sigf(float v) { return 1.f / (1.f + __expf(-v)); }
__device__ __forceinline__ float clipp(float p) {
    return fminf(fmaxf(p, 1e-12f), 1.f - 1e-12f);
}

// ---------------- kernel 1: zero / init workspace ----------------
__global__ void yolo_init(float* obj, float* noobj, float* tx, float* ty,
                          float* tw, float* th, unsigned* clsbits, float* accum) {
    int i = blockIdx.x * blockDim.x + threadIdx.x;
    if (i < CELLS_) {
        obj[i] = 0.f;  noobj[i] = 1.f;
        tx[i] = 0.f;   ty[i] = 0.f;  tw[i] = 0.f;  th[i] = 0.f;
        clsbits[3 * i + 0] = 0u; clsbits[3 * i + 1] = 0u; clsbits[3 * i + 2] = 0u;
    }
    if (i < 8) accum[i] = 0.f;
}

// ---------------- kernel 2: sparse target build (256 targets) ----------------
__global__ void yolo_build(const float* __restrict__ tgt, float* obj, float* noobj,
                           float* tx, float* ty, float* tw, float* th,
                           unsigned* clsbits) {
    int n = threadIdx.x;
    if (n >= NT_) return;
    const float* t = tgt + n * 6;
    int   img = (int)t[0];
    float gxc = t[1] * (float)W_, gyc = t[2] * (float)H_;
    float gw  = t[3] * (float)W_, gh  = t[4] * (float)H_;
    int   label = (int)t[5];

    float ious[3];
    float best_iou = -1.f; int best = 0;
    for (int a = 0; a < 3; ++a) {
        float aw = d_aw[a], ah = d_ah[a];
        float inter = fminf(aw, gw) * fminf(ah, gh);
        float iou = inter / (aw * ah + 1e-16f + gw * gh - inter);
        ious[a] = iou;
        if (iou > best_iou) { best_iou = iou; best = a; }   // first-max wins (argmax)
    }
    int gi = (int)gxc, gj = (int)gyc;
    int cell = ((img * A_ + best) * H_ + gj) * W_ + gi;

    obj[cell] = 1.f;                                  // duplicates: same value
    tx[cell]  = gxc - floorf(gxc);                    // duplicates: last-write-wins
    ty[cell]  = gyc - floorf(gyc);
    tw[cell]  = __logf(gw / d_aw[best] + 1e-16f);
    th[cell]  = __logf(gh / d_ah[best] + 1e-16f);
    atomicOr(&clsbits[cell * 3 + (label >> 5)], 1u << (label & 31));

    // noobj zeroed where a==best OR iou > 0.5
    for (int a = 0; a < 3; ++a) {
        if (a == best || ious[a] > 0.5f) {
            int c2 = ((img * A_ + a) * H_ + gj) * W_ + gi;
            noobj[c2] = 0.f;
        }
    }
}

// ---------------- kernel 3: streaming transform + loss partials ----------------
__global__ __launch_bounds__(TILE_)
void yolo_main(const float* __restrict__ x, float* __restrict__ out,
               const float* __restrict__ obj, const float* __restrict__ noobj,
               const float* __restrict__ tx, const float* __restrict__ ty,
               const float* __restrict__ tw, const float* __restrict__ th,
               const unsigned* __restrict__ clsbits, float* accum) {
    __shared__ __align__(16) float    tile[CH_ * ROW_];   // [ch][cell] padded rows
    __shared__ __align__(16) float    objs[TILE_];
    __shared__ __align__(16) unsigned clsb[TILE_ * 3];
    __shared__ float red[TILE_ / 32][8];

    const int tid   = threadIdx.x;
    const int plane = blockIdx.y;                  // b*A + a
    const int b     = plane / A_;
    const int a     = plane - b * A_;
    const int cell0 = blockIdx.x * TILE_;
    const int count = (HW_ - cell0 < TILE_) ? (HW_ - cell0) : TILE_;  // 128 or 16
    const int planeCellBase = plane * HW_ + cell0;
    const int full  = (count == TILE_);
    const int rowstride = full ? ROW_ : count;     // TDM pad only for full rows

    // --- TDM: one TENSOR_LOAD_TO_LDS moves the whole 85 x count f32 tile ---
    // D# built per CDNA5 ISA (2-D tensor, groups 2/3 zero).
    if (tid < 32) {                                // wave 0 issues; TDM ignores EXEC
        uint64_t ga = (uint64_t)(uintptr_t)(x + (size_t)plane * CH_ * HW_ + cell0);
        v4u g0;
        g0[0] = 1u;                                         // count=1, user mode
        g0[1] = (unsigned)(uintptr_t)&tile[0];              // lds_addr (bytes)
        g0[2] = (unsigned)ga;                               // global_addr[31:0]
        g0[3] = (unsigned)((ga >> 32) & 0x01FFFFFFu) | (2u << 30);  // addr[56:32], type=2
        v8i g1;
        unsigned d0 = (2u << 16);                           // data_size=4B
        if (full) d0 |= (1u << 20) | (6u << 22) | (3u << 25); // pad 4 DW per 128 DW
        g1[0] = (int)d0;
        g1[1] = (int)(((unsigned)HW_ & 0xFFFFu) << 16);     // tensor_dim0[15:0]
        g1[2] = (int)((((unsigned)HW_ >> 16) & 0xFFFFu) | (((unsigned)CH_) << 16)); // dim0 hi, dim1 lo
        g1[3] = (int)(((unsigned)count) << 16);             // dim1 hi=0, tile_dim0=count
        g1[4] = (int)CH_;                                   // tile_dim1=85, tile_dim2=0
        g1[5] = (int)HW_;                                   // tensor_dim0_stride[31:0]
        g1[6] = 0;                                          // stride hi, dim1_stride lo
        g1[7] = 0;
        v4i g2 = {0, 0, 0, 0}, g3 = {0, 0, 0, 0};
        v8i g4 = {0, 0, 0, 0, 0, 0, 0, 0};
        __builtin_amdgcn_tensor_load_to_lds(g0, g1, g2, g3, g4, 0);
        __builtin_amdgcn_s_wait_tensorcnt(0);
    }
    // obj-mask tile (reused ~83x/cell) and class-bit tile (80x/cell): async DMA
    if (tid < 32 && (tid << 2) < count) {
        const float* g = obj + planeCellBase + (tid << 2);
        unsigned laddr = (unsigned)(uintptr_t)&objs[tid << 2];
        asm volatile("global_load_async_to_lds_b128 %0, %1, off"
                     :: "v"(laddr), "v"(g) : "memory");
    }
    if (tid < 96 && (tid << 2) < 3 * count) {
        const unsigned* g = clsbits + (size_t)planeCellBase * 3 + (tid << 2);
        unsigned laddr = (unsigned)(uintptr_t)&clsb[tid << 2];
        asm volatile("global_load_async_to_lds_b128 %0, %1, off"
                     :: "v"(laddr), "v"(g) : "memory");
    }
    asm volatile("s_wait_asynccnt 0" ::: "memory");
    __syncthreads();                               // releases tile to all 4 waves

    float axy = 0.f, awh = 0.f, acobj = 0.f, acnoobj = 0.f, acls = 0.f;
    float anobj = 0.f, annoobj = 0.f;

    const size_t outBase = (size_t)(b * (A_ * HW_) + a * HW_ + cell0) * CH_;
    const int    total = count * CH_;

    for (int o = tid; o < total; o += TILE_) {       // fully coalesced NT stores
        int cell = o / CH_;
        int ch   = o - cell * CH_;
        float v  = tile[ch * rowstride + cell];

        int gcell = cell0 + cell;
        int gy = gcell / W_;
        int gx = gcell - gy * W_;

        float sv = sigf(v);
        float outv;
        if      (ch == 0) outv = (sv + (float)gx) * STRIDE_;   // bx*stride_w
        else if (ch == 1) outv = (sv + (float)gy) * STRIDE_;   // by*stride_h
        else if (ch == 2) outv = __expf(v) * d_aw[a];          // bw*stride_w = exp(pw)*aw
        else if (ch == 3) outv = __expf(v) * d_ah[a];
        else              outv = sv;                           // conf / cls sigmoid
        __builtin_nontemporal_store(outv, &out[outBase + o]);  // streamed, no reuse

        int ca = planeCellBase + cell;
        if (ch == 0) {                    // loss_x and loss_y both use sigmoid(cx)
            float m = objs[cell];
            if (m != 0.f) {
                float dx = sv - tx[ca], dy = sv - ty[ca];
                axy += dx * dx + dy * dy;
            }
        } else if (ch == 2) {             // loss_w and loss_h both use pw
            float m = objs[cell];
            if (m != 0.f) {
                float dw = v - tw[ca], dh = v - th[ca];
                awh += dw * dw + dh * dh;
            }
        } else if (ch == 4) {             // conf BCE: obj cells t=1, noobj cells t=0
            float mo = objs[cell], mn = noobj[ca];
            float p = clipp(sv);
            acobj   += mo * (-__logf(p));
            acnoobj += mn * (-__logf(1.f - p));
            anobj   += mo;
            annoobj += mn;
        } else if (ch >= 5) {             // class BCE under obj mask
            float m = objs[cell];
            if (m != 0.f) {
                int c = ch - 5;
                unsigned bit = (clsb[cell * 3 + (c >> 5)] >> (c & 31)) & 1u;
                float p = clipp(sv);
                acls += bit ? -__logf(p) : -__logf(1.f - p);
            }
        }
    }

    // --- wave32 shuffle reduction, then cross-wave via LDS, then f32 atomics ---
    float acc[7] = {axy, awh, acobj, acnoobj, acls, anobj, annoobj};
    int lane = tid & 31, wv = tid >> 5;
    for (int k = 0; k < 7; ++k) {
        float v = acc[k];
        for (int off = 16; off; off >>= 1) v += __shfl_xor(v, off, 32);
        acc[k] = v;
    }
    if (lane == 0)
        for (int k = 0; k < 7; ++k) red[wv][k] = acc[k];
    __syncthreads();
    if (tid == 0) {
        for (int k = 0; k < 7; ++k) {
            float s = 0.f;
            for (int w = 0; w < TILE_ / 32; ++w) s += red[w][k];
            atomicAdd(&accum[k], s);
        }
    }
}

// ---------------- kernel 4: finalize scalar total ----------------
__global__ void yolo_fin(const float* accum, float* out, int out_size) {
    if (threadIdx.x == 0 && blockIdx.x == 0) {
        float nobj   = fmaxf(accum[5], 1.f);
        float nnoobj = fmaxf(accum[6], 1.f);
        float ncls   = fmaxf(80.f * accum[5], 1.f);
        float tot = accum[0] / nobj            // loss_x + loss_y
                  + accum[1] / nobj            // loss_w + loss_h
                  + accum[2] / nobj            // OBJ_SCALE * conf_obj
                  + 0.5f * accum[3] / nnoobj   // NOOBJ_SCALE * conf_noobj
                  + accum[4] / ncls;           // loss_cls
        out[out_size - 1] = tot;
    }
}

extern "C" void kernel_launch(void* const* d_in, const int* in_sizes, int n_in,
                              void* d_out, int out_size, void* d_ws, size_t ws_size,
                              hipStream_t stream) {
    const float* x   = (const float*)d_in[0];
    const float* tgt = (const float*)d_in[1];
    float* out = (float*)d_out;

    float*    ws      = (float*)d_ws;
    float*    obj     = ws;
    float*    noobj   = obj + CELLS_;
    float*    tx      = noobj + CELLS_;
    float*    ty      = tx + CELLS_;
    float*    tw      = ty + CELLS_;
    float*    th      = tw + CELLS_;
    unsigned* clsbits = (unsigned*)(th + CELLS_);       // 3 * CELLS_ u32
    float*    accum   = (float*)(clsbits + 3 * CELLS_); // 8 floats

    yolo_init<<<(CELLS_ + 255) / 256, 256, 0, stream>>>(obj, noobj, tx, ty, tw, th,
                                                        clsbits, accum);
    yolo_build<<<1, 256, 0, stream>>>(tgt, obj, noobj, tx, ty, tw, th, clsbits);

    dim3 grid((HW_ + TILE_ - 1) / TILE_, B_ * A_);
    yolo_main<<<grid, TILE_, 0, stream>>>(x, out, obj, noobj, tx, ty, tw, th,
                                          clsbits, accum);
    yolo_fin<<<1, 1, 0, stream>>>(accum, out, out_size);
}